// RuleMLP_46643344835133
// MI455X (gfx1250) — compile-verified
//
#include <hip/hip_runtime.h>
#include <hip/hip_bf16.h>

// Sizes from the reference
#define BB   8192   // batch
#define FF   2048   // features of x
#define NN   512    // rules
#define GG   16     // group size
#define NG   (NN*GG)        // 8192 columns of h
#define EPSF 1e-5f

typedef __attribute__((ext_vector_type(2))) float v2f;
typedef __attribute__((ext_vector_type(8))) float v8f;

// Workspace layout (float offsets)
#define OFF_SUM    0           // NG : column sums of raw h
#define OFF_SQ     8192        // NG : column sum-of-squares of raw h
#define OFF_W2EFF  16384       // NG : rs * bn1_w * W2
#define OFF_OCONST 24576       // NN : per-rule constant for o
#define OFF_OSUM   25088       // NN
#define OFF_OSQ    25600       // NN
#define OFF_WEFF   26112       // NN : rs_o * fc2_w
#define OFF_CEFF   26624       // 1
#define OFF_HDR    27136       // header size (zeroed)
#define OFF_O      27136       // BB*NN raw o values, layout o[n*BB + b]

__global__ __launch_bounds__(256) void k_init(float* ws) {
    int i = blockIdx.x * 256 + threadIdx.x;
    if (i < OFF_HDR) ws[i] = 0.0f;
}

// Gather one 16x4-per-lane A-fragment set (8 scattered dwords) for a tile.
// Unsigned 32-bit offsets -> SADDR + voffset global loads (x is 64MB < 4GB).
__device__ __forceinline__ void gather4(const float* __restrict__ x, unsigned rowoff,
                                        const int* s_idx, int khalf, v2f a[4]) {
#pragma unroll
    for (int kk = 0; kk < 4; ++kk) {
        int k0 = kk*4 + khalf;
        a[kk].x = x[rowoff + (unsigned)s_idx[k0]];
        a[kk].y = x[rowoff + (unsigned)s_idx[k0 + 1]];
    }
}

// 16x16 fp32 tile via chained V_WMMA_F32_16X16X4_F32 (K = 4*4 = 16)
__device__ __forceinline__ v8f wchain(const v2f a[4], const v2f bf[4]) {
    v8f c = {0.f,0.f,0.f,0.f,0.f,0.f,0.f,0.f};
#pragma unroll
    for (int kk = 0; kk < 4; ++kk)
        c = __builtin_amdgcn_wmma_f32_16x16x4_f32(
                false, a[kk], false, bf[kk], (short)0, c, false, false);
    return c;
}

// ---------------- Pass 1: WMMA GEMM -> per-column sum / sumsq ----------------
// grid (NN, 4), block 256 (8 waves). Each block: one rule n, 2048 batch rows.
// Each wave handles 16 tiles of 16 rows. h_raw tile = A(g gather 16x16) x B(W1^T).
__global__ __launch_bounds__(256) void k_stats(const float* __restrict__ x,
                                               const int*   __restrict__ idx,
                                               const float* __restrict__ W1,
                                               float* __restrict__ gsum,
                                               float* __restrict__ gsq) {
    __shared__ int   s_idx[GG];
    __shared__ float s_w1[GG*GG];
    const int n      = blockIdx.x;
    const int b_base = blockIdx.y * 2048;
    const int tid    = threadIdx.x;
    if (tid < GG) s_idx[tid] = idx[n*GG + tid];
    s_w1[tid] = W1[n*GG*GG + tid];
    __syncthreads();

    const int lane  = tid & 31;
    const int wave  = tid >> 5;
    const int col   = lane & 15;           // A row index == B/D column index
    const int khalf = (lane >> 4) << 1;    // lanes 16-31 hold K+2,K+3

    // B fragments: Bmat[k][col] = W1[n][col][k] (transposed use of W1)
    v2f bf[4];
#pragma unroll
    for (int kk = 0; kk < 4; ++kk) {
        int k0 = kk*4 + khalf;
        bf[kk].x = s_w1[col*GG + k0];
        bf[kk].y = s_w1[col*GG + k0 + 1];
    }

    // vector accumulators: horizontal reduction deferred to after the loop
    v8f vs = {0.f,0.f,0.f,0.f,0.f,0.f,0.f,0.f};
    v8f vq = {0.f,0.f,0.f,0.f,0.f,0.f,0.f,0.f};

    // double-buffered gather: issue tile i+1's loads before tile i's WMMA chain
    unsigned rowoff   = (unsigned)(b_base + wave*16 + col) * FF;
    const unsigned dr = 128u * FF;              // 8 waves * 16 rows per step
    v2f a[4];
    gather4(x, rowoff, s_idx, khalf, a);
    for (int i = 0; i < 15; ++i) {
        v2f an[4];
        gather4(x, rowoff + dr, s_idx, khalf, an);
        v8f c = wchain(a, bf);
        vs += c;
        vq += c * c;
#pragma unroll
        for (int kk = 0; kk < 4; ++kk) a[kk] = an[kk];
        rowoff += dr;
    }
    {
        v8f c = wchain(a, bf);
        vs += c;
        vq += c * c;
    }

    // horizontal 8->1 reduction, once
    float s = 0.f, s2 = 0.f;
#pragma unroll
    for (int r = 0; r < 8; ++r) { s += vs[r]; s2 += vq[r]; }
    // combine the two row-halves of each column
    s  += __shfl_xor(s,  16, 32);
    s2 += __shfl_xor(s2, 16, 32);
    if (lane < 16) {
        atomicAdd(&gsum[n*GG + lane], s);
        atomicAdd(&gsq [n*GG + lane], s2);
    }
}

// ---------------- finalize h stats: w2eff, o_const ----------------
// b1 cancels inside batchnorm (shift invariance), so it is never needed.
__global__ void k_fin_h(const float* __restrict__ gsum, const float* __restrict__ gsq,
                        const float* __restrict__ bn1_w, const float* __restrict__ bn1_b,
                        const float* __restrict__ W2, const float* __restrict__ b2,
                        float* __restrict__ w2eff, float* __restrict__ oconst) {
    int n = blockIdx.x, h = threadIdx.x;      // blockDim = 16
    int i = n*GG + h;
    float mean = gsum[i] * (1.0f/BB);
    float var  = gsq[i]  * (1.0f/BB) - mean*mean;
    float rs   = rsqrtf(var + EPSF);
    float w    = bn1_w[i] * rs;
    float w2   = W2[i];
    w2eff[i]   = w * w2;
    float contrib = (bn1_b[i] - mean * w) * w2;
#pragma unroll
    for (int m = 1; m < 16; m <<= 1) contrib += __shfl_xor(contrib, m, 32);
    if (h == 0) oconst[n] = b2[n] + contrib;
}

// ---------------- Pass 2: recompute h tiles, contract -> o, o stats ----------
__global__ __launch_bounds__(256) void k_opass(const float* __restrict__ x,
                                               const int*   __restrict__ idx,
                                               const float* __restrict__ W1,
                                               const float* __restrict__ w2eff,
                                               const float* __restrict__ oconst,
                                               float* __restrict__ o_out,
                                               float* __restrict__ osum,
                                               float* __restrict__ osq) {
    __shared__ int   s_idx[GG];
    __shared__ float s_w1[GG*GG];
    __shared__ float s_w2[GG];
    __shared__ float s_tile[8][16*17];         // per-wave D tile, padded
    const int n      = blockIdx.x;
    const int b_base = blockIdx.y * 2048;
    const int tid    = threadIdx.x;
    if (tid < GG) { s_idx[tid] = idx[n*GG + tid]; s_w2[tid] = w2eff[n*GG + tid]; }
    s_w1[tid] = W1[n*GG*GG + tid];
    __syncthreads();

    const float oc    = oconst[n];
    const int lane    = tid & 31;
    const int wave    = tid >> 5;
    const int col     = lane & 15;
    const int khalf   = (lane >> 4) << 1;
    const int rowhalf = (lane >> 4) << 3;      // 0 or 8

    v2f bf[4];
#pragma unroll
    for (int kk = 0; kk < 4; ++kk) {
        int k0 = kk*4 + khalf;
        bf[kk].x = s_w1[col*GG + k0];
        bf[kk].y = s_w1[col*GG + k0 + 1];
    }

    float s = 0.f, s2 = 0.f;
    unsigned rowoff   = (unsigned)(b_base + wave*16 + col) * FF;
    const unsigned dr = 128u * FF;
    v2f a[4];
    gather4(x, rowoff, s_idx, khalf, a);
    for (int i = 0; i < 16; ++i) {
        v2f an[4];
        if (i < 15) gather4(x, rowoff + dr, s_idx, khalf, an);
        v8f c = wchain(a, bf);
        // stage tile through LDS to form per-row dots (same-wave DS ops are in order)
#pragma unroll
        for (int r = 0; r < 8; ++r) s_tile[wave][(rowhalf + r)*17 + col] = c[r];
        __builtin_amdgcn_wave_barrier();
        if (lane < 16) {
            float acc = oc;
#pragma unroll
            for (int hh = 0; hh < 16; ++hh) acc += s_tile[wave][lane*17 + hh] * s_w2[hh];
            int b = b_base + (wave + (i << 3))*16 + lane;
            o_out[n*BB + b] = acc;             // coalesced: lanes -> consecutive b
            s  += acc;
            s2 += acc*acc;
        }
        __builtin_amdgcn_wave_barrier();
#pragma unroll
        for (int kk = 0; kk < 4; ++kk) a[kk] = an[kk];
        rowoff += dr;
    }
    // full-wave reduction (lanes 16-31 carry zeros)
#pragma unroll
    for (int m = 1; m < 32; m <<= 1) { s += __shfl_xor(s, m, 32); s2 += __shfl_xor(s2, m, 32); }
    if (lane == 0) { atomicAdd(&osum[n], s); atomicAdd(&osq[n], s2); }
}

// ---------------- finalize o stats ----------------
__global__ __launch_bounds__(512) void k_fin_o(const float* __restrict__ osum,
                                               const float* __restrict__ osq,
                                               const float* __restrict__ fc2_w,
                                               const float* __restrict__ fc2_b,
                                               float* __restrict__ weff,
                                               float* __restrict__ ceff) {
    __shared__ float red[16];
    int n = threadIdx.x;                       // 512 threads, 16 waves
    float mean = osum[n] * (1.0f/BB);
    float var  = osq[n]  * (1.0f/BB) - mean*mean;
    float rs   = rsqrtf(var + EPSF);
    float w    = rs * fc2_w[n];
    weff[n]    = w;
    float contrib = -mean * w;
#pragma unroll
    for (int m = 1; m < 32; m <<= 1) contrib += __shfl_xor(contrib, m, 32);
    if ((n & 31) == 0) red[n >> 5] = contrib;
    __syncthreads();
    if (n == 0) {
        float tot = fc2_b[0];
        for (int i = 0; i < 16; ++i) tot += red[i];
        *ceff = tot;
    }
}

// ---------------- final projection: out[b] = ceff + sum_n o[n][b]*weff[n] ----
__global__ __launch_bounds__(256) void k_out(const float* __restrict__ o,
                                             const float* __restrict__ weff,
                                             const float* __restrict__ ceff,
                                             float* __restrict__ out) {
    __shared__ float s_w[NN];
    int tid = threadIdx.x;
    s_w[tid]       = weff[tid];
    s_w[tid + 256] = weff[tid + 256];
    __syncthreads();
    unsigned b = blockIdx.x * 256u + tid;
    float acc = *ceff;
#pragma unroll 8
    for (unsigned n = 0; n < NN; ++n) acc += o[n*BB + b] * s_w[n];  // coalesced per n
    out[b] = acc;
}

extern "C" void kernel_launch(void* const* d_in, const int* in_sizes, int n_in,
                              void* d_out, int out_size, void* d_ws, size_t ws_size,
                              hipStream_t stream) {
    (void)in_sizes; (void)n_in; (void)out_size; (void)ws_size;
    const float* x     = (const float*)d_in[0];
    const int*   idx   = (const int*)  d_in[1];
    const float* W1    = (const float*)d_in[2];
    /* d_in[3] = b1 : unused — cancels inside batchnorm (shift invariance) */
    const float* bn1_w = (const float*)d_in[4];
    const float* bn1_b = (const float*)d_in[5];
    const float* W2    = (const float*)d_in[6];
    const float* b2    = (const float*)d_in[7];
    const float* fc2_w = (const float*)d_in[8];
    const float* fc2_b = (const float*)d_in[9];
    float* ws  = (float*)d_ws;
    float* out = (float*)d_out;

    k_init <<<(OFF_HDR + 255)/256, 256, 0, stream>>>(ws);
    k_stats<<<dim3(NN, 4), 256, 0, stream>>>(x, idx, W1, ws + OFF_SUM, ws + OFF_SQ);
    k_fin_h<<<NN, GG, 0, stream>>>(ws + OFF_SUM, ws + OFF_SQ, bn1_w, bn1_b, W2, b2,
                                   ws + OFF_W2EFF, ws + OFF_OCONST);
    k_opass<<<dim3(NN, 4), 256, 0, stream>>>(x, idx, W1, ws + OFF_W2EFF, ws + OFF_OCONST,
                                             ws + OFF_O, ws + OFF_OSUM, ws + OFF_OSQ);
    k_fin_o<<<1, NN, 0, stream>>>(ws + OFF_OSUM, ws + OFF_OSQ, fc2_w, fc2_b,
                                  ws + OFF_WEFF, ws + OFF_CEFF);
    k_out  <<<BB/256, 256, 0, stream>>>(ws + OFF_O, ws + OFF_WEFF, ws + OFF_CEFF, out);
}